// SimpleAttention_36146444763805
// MI455X (gfx1250) — compile-verified
//
#include <hip/hip_runtime.h>
#include <math.h>

// ---------------------------------------------------------------------------
// SimpleAttention (graph segment-softmax) for MI455X / gfx1250.
//
// out[n,:] = (sum_{e: self[e]==n} exp(q[self]·k[nbr]/8) / denom[n]) * v[n,:]
//
// Pass 2 computes the 800k edge dot-products with V_WMMA_F32_16X16X4_F32:
// one wave32 = 16 edges, A = 16 gathered q rows, B = 16 gathered k rows
// (transposed), 16 chained K=4 WMMAs, scores = diagonal of the 16x16 D tile.
// Per K-chunk each lane issues one global_load_b64 from its q row and one
// from its k row -> exactly the minimal 512 B/edge of gather traffic, all
// served from the 192 MB L2 (q+k+v = 38 MB).
// ---------------------------------------------------------------------------

typedef __attribute__((ext_vector_type(2))) float v2f;
typedef __attribute__((ext_vector_type(8))) float v8f;

#define DVEC 64  // feature dim

__global__ __launch_bounds__(256) void sa_zero_f32(float* __restrict__ p, int n) {
    int i = blockIdx.x * blockDim.x + threadIdx.x;
    if (i < n) p[i] = 0.0f;
}

// One wave per 16 edges. EXEC stays all-ones through the WMMA region
// (tail edges are clamped to a valid index; their results are masked later).
__global__ __launch_bounds__(256) void sa_edge_scores_wmma(
    const float* __restrict__ q, const float* __restrict__ k,
    const int* __restrict__ self_idx, const int* __restrict__ nbr_idx,
    float* __restrict__ ex_buf, float* __restrict__ denom, int E)
{
    const int lane = threadIdx.x & 31;
    const int m    = lane & 15;     // edge slot within the 16-edge tile
    const int half = lane >> 4;     // 0: K pair {4c,4c+1}; 1: K pair {4c+2,4c+3}

    const long wave   = (long)blockIdx.x * (blockDim.x >> 5) + (threadIdx.x >> 5);
    const int  e_base = (int)(wave * 16);
    const int  e      = e_base + m;
    const int  ec     = e < E ? e : E - 1;  // clamp for safe loads on tail

    const int row = self_idx[ec];
    const int nb  = nbr_idx[ec];

    // A (16x4 f32): lane m holds q[row_m][4c + 2*half + {0,1}] in {V0,V1}.
    // B (4x16 f32): lane m holds k[nb_m ][4c + 2*half + {0,1}] in {V0,V1}.
    const float* __restrict__ qp = q + (long)row * DVEC + half * 2;
    const float* __restrict__ kp = k + (long)nb  * DVEC + half * 2;

    v8f acc = {};
#pragma unroll
    for (int c = 0; c < 16; ++c) {
        v2f a = *(const v2f*)(qp + c * 4);   // global_load_b64
        v2f b = *(const v2f*)(kp + c * 4);   // global_load_b64
        // v_wmma_f32_16x16x4_f32: D = A x B + C (accumulate over K=64)
        acc = __builtin_amdgcn_wmma_f32_16x16x4_f32(
            /*neg_a=*/false, a, /*neg_b=*/false, b,
            /*c_mod=*/(short)0, acc, /*reuse_a=*/false, /*reuse_b=*/false);
    }

    // Diagonal extraction from the 16x16 f32 C/D layout:
    //   D(i,i), i<8  -> lane i,     VGPR i
    //   D(i,i), i>=8 -> lane 16+i,  VGPR i-8   (lanes 24..31, vgpr lane-24)
    const int  dl     = (lane < 8) ? lane : (lane - 24);
    const bool active = (lane < 8) || (lane >= 24);

    float s = acc[0];
    s = (dl == 1) ? acc[1] : s;
    s = (dl == 2) ? acc[2] : s;
    s = (dl == 3) ? acc[3] : s;
    s = (dl == 4) ? acc[4] : s;
    s = (dl == 5) ? acc[5] : s;
    s = (dl == 6) ? acc[6] : s;
    s = (dl == 7) ? acc[7] : s;

    if (active && e < E) {
        // scores ~ N(0,1): exp never overflows
        float ex = __expf(s * 0.125f);   // 1/sqrt(64)
        ex_buf[e] = ex;
        atomicAdd(&denom[row], ex);      // global_atomic_add_f32 (L2 atomic)
    }
}

// Per-edge: softmax weight, accumulated as a SCALAR per node (v is gathered
// by self_indices in the reference, so out[n] = (sum attn) * v[n]).
__global__ __launch_bounds__(256) void sa_edge_scatter(
    const float* __restrict__ ex_buf, const float* __restrict__ denom,
    const int* __restrict__ self_idx, float* __restrict__ accA, int E)
{
    int e = blockIdx.x * blockDim.x + threadIdx.x;
    if (e < E) {
        int   rw   = self_idx[e];
        float attn = ex_buf[e] / denom[rw];
        atomicAdd(&accA[rw], attn);
    }
}

// out[n,:] = accA[n] * v[n,:], vectorized as float4 (16 float4 per node row).
__global__ __launch_bounds__(256) void sa_finalize(
    const float* __restrict__ accA, const float* __restrict__ v,
    float* __restrict__ out, int n4)
{
    int i = blockIdx.x * blockDim.x + threadIdx.x;
    if (i < n4) {
        float4 vv = ((const float4*)v)[i];
        float  a  = accA[i >> 4];          // 64 floats = 16 float4 per node
        float4 o;
        o.x = a * vv.x; o.y = a * vv.y; o.z = a * vv.z; o.w = a * vv.w;
        ((float4*)out)[i] = o;
    }
}

extern "C" void kernel_launch(void* const* d_in, const int* in_sizes, int n_in,
                              void* d_out, int out_size, void* d_ws, size_t ws_size,
                              hipStream_t stream) {
    const float* q   = (const float*)d_in[0];
    const float* k   = (const float*)d_in[1];
    const float* v   = (const float*)d_in[2];
    const int*   si  = (const int*)d_in[3];
    const int*   ni  = (const int*)d_in[4];
    float*       out = (float*)d_out;

    const int N = in_sizes[0] / DVEC;   // 50000
    const int E = in_sizes[3];          // 800000

    // workspace layout: denom[N] | accA[N] | ex[E]
    float* denom = (float*)d_ws;
    float* accA  = denom + N;
    float* exb   = accA + N;

    // 1) zero the two accumulators (contiguous 2N floats)
    {
        int n = 2 * N;
        sa_zero_f32<<<(n + 255) / 256, 256, 0, stream>>>(denom, n);
    }
    // 2) WMMA edge scores + exp + denom scatter (8 waves/block = 128 edges/block)
    {
        int blocks = (E + 127) / 128;
        sa_edge_scores_wmma<<<blocks, 256, 0, stream>>>(q, k, si, ni, exb, denom, E);
    }
    // 3) per-edge softmax weight -> per-node scalar mass
    {
        sa_edge_scatter<<<(E + 255) / 256, 256, 0, stream>>>(exb, denom, si, accA, E);
    }
    // 4) broadcast-scale v rows
    {
        int n4 = N * (DVEC / 4);
        sa_finalize<<<(n4 + 255) / 256, 256, 0, stream>>>(accA, v, out, n4);
    }
}